// LogicNetwork_84267258347625
// MI455X (gfx1250) — compile-verified
//
#include <hip/hip_runtime.h>

// LogicNetwork fused kernel for MI455X (gfx1250, wave32).
// One thread = one batch element for the attention pipeline (fp32, registers).
// Final ActionLayer GEMM (flat[B,12] @ act_W.T[12,9]) done with
// V_WMMA_F32_16X16X4_F32: M=16 batch rows per tile, N=16 (9 valid), K=12 as 3x K=4.
// All global I/O is staged through LDS for fully-coalesced transactions;
// bounds checks are block-uniform so the hot path has no divergent guards.

typedef __attribute__((ext_vector_type(2))) float v2f;
typedef __attribute__((ext_vector_type(8))) float v8f;

#define BLK 256
#define FSTRIDE 13  // LDS row stride for 12 features (coprime with 64 banks)

__device__ __forceinline__ float clamp01(float x) { return fminf(fmaxf(x, 0.f), 1.f); }

__global__ __launch_bounds__(BLK) void logic_net_wmma_kernel(
    const float* __restrict__ state,        // [B,18]
    const float* __restrict__ c_templates,  // [12,2]
    const float* __restrict__ c_gammas,     // [12,2]
    const float* __restrict__ a_templates,  // [6,2,2]
    const float* __restrict__ a_gammas,     // [6,2,2]
    const float* __restrict__ a_body_W,     // [6,2,4,2]
    const float* __restrict__ a_body_b,     // [6,2,4]
    const float* __restrict__ a_head_W,     // [6,2,4]
    const float* __restrict__ a_head_b,     // [6,2]
    const float* __restrict__ act_W,        // [9,12]
    const float* __restrict__ act_b,        // [9]
    float* __restrict__ out,                // [B,9]
    int Bn)
{
    __shared__ float s_state[BLK * 18];       // coalesced staging of inputs
    __shared__ float s_flat[BLK * FSTRIDE];   // 12 features per row -> WMMA A tiles
    __shared__ float s_q[BLK * 9];            // WMMA D results -> coalesced stores

    const int t = threadIdx.x;
    const size_t blockBase = (size_t)blockIdx.x * BLK;
    const bool fullBlock = (blockBase + BLK) <= (size_t)Bn;  // wave-uniform

    // ---- Stage 256x18 floats with fully coalesced loads ----
    const float* gin = state + blockBase * 18;
    if (fullBlock) {
#pragma unroll
        for (int k = 0; k < 18; ++k)
            s_state[k * BLK + t] = gin[(size_t)k * BLK + t];
    } else {
        const size_t remain = (size_t)Bn * 18 - blockBase * 18;
#pragma unroll
        for (int k = 0; k < 18; ++k) {
            size_t idx = (size_t)k * BLK + t;
            s_state[k * BLK + t] = (idx < remain) ? gin[idx] : 0.f;
        }
    }
    __syncthreads();

    // board in registers: bd[2*i+p] = board[i][p]
    float bd[18];
    {
        const float* my = &s_state[t * 18];
#pragma unroll
        for (int k = 0; k < 18; ++k) bd[k] = my[k];
    }

    // ---- ConcreteLayer: 12 rules, softmax-attention over 9 squares ----
    float cf0[12], cf1[12];
#pragma unroll
    for (int r = 0; r < 12; ++r) {
        const float t0 = c_templates[2 * r + 0];
        const float t1 = c_templates[2 * r + 1];
        const float w0 = 1.f - clamp01(c_gammas[2 * r + 0]);
        const float w1 = 1.f - clamp01(c_gammas[2 * r + 1]);
        float ms[9];
        float m = 3.4e38f;
#pragma unroll
        for (int i = 0; i < 9; ++i) {
            float d0 = t0 - bd[2 * i + 0];
            float d1 = t1 - bd[2 * i + 1];
            ms[i] = w0 * d0 * d0 + w1 * d1 * d1;
            m = fminf(m, ms[i]);
        }
        float sum = 0.f, match = 0.f;
#pragma unroll
        for (int i = 0; i < 9; ++i) {
            float e = __expf(m - ms[i]);   // softmax(-ms) with max-subtraction
            sum += e;
            match += e * bd[2 * i + 1];
        }
        cf0[r] = __expf(-m);               // pattern strength = exp(-min ms)
        cf1[r] = match / sum;              // matched[...,1]
    }

    // ---- AbstractionLayer: 6 rules x 2 clauses, softmax over 12 concrete feats ----
    float flatv[12];
#pragma unroll
    for (int r = 0; r < 6; ++r) {
        float cap[4] = {0.f, 0.f, 0.f, 0.f};
#pragma unroll
        for (int j = 0; j < 2; ++j) {
            const int rj = r * 2 + j;
            const float t0 = a_templates[2 * rj + 0];
            const float t1 = a_templates[2 * rj + 1];
            const float w0 = 1.f - clamp01(a_gammas[2 * rj + 0]);
            const float w1 = 1.f - clamp01(a_gammas[2 * rj + 1]);
            float ms[12];
            float m = 3.4e38f;
#pragma unroll
            for (int i = 0; i < 12; ++i) {
                float d0 = t0 - cf0[i];
                float d1 = t1 - cf1[i];
                ms[i] = w0 * d0 * d0 + w1 * d1 * d1;
                m = fminf(m, ms[i]);
            }
            float sum = 0.f, s0 = 0.f, s1 = 0.f;
#pragma unroll
            for (int i = 0; i < 12; ++i) {
                float e = __expf(m - ms[i]);
                sum += e;
                s0 += e * cf0[i];
                s1 += e * cf1[i];
            }
            const float inv = 1.f / sum;
            s0 *= inv;
            s1 *= inv;
#pragma unroll
            for (int v = 0; v < 4; ++v) {
                const int bvi = rj * 4 + v;
                cap[v] += fmaf(s0, a_body_W[bvi * 2 + 0],
                          fmaf(s1, a_body_W[bvi * 2 + 1], a_body_b[bvi]));
            }
        }
#pragma unroll
        for (int l = 0; l < 2; ++l) {
            float acc = a_head_b[2 * r + l];
#pragma unroll
            for (int v = 0; v < 4; ++v)
                acc = fmaf(cap[v], a_head_W[(2 * r + l) * 4 + v], acc);
            flatv[2 * r + l] = acc;
        }
    }

    // ---- Stage features into LDS for WMMA A-matrix gather ----
#pragma unroll
    for (int k = 0; k < 12; ++k) s_flat[t * FSTRIDE + k] = flatv[k];
    __syncthreads();

    // ---- ActionLayer via V_WMMA_F32_16X16X4_F32 ----
    // q[b,n] = sum_k flat[b,k] * act_W[n,k] + act_b[n]; K=12 -> 3 chunks of 4.
    const int lane = t & 31;
    const int wv   = t >> 5;
    const int n    = lane & 15;   // N column (action), valid for n<9
    const int kh   = lane >> 4;   // lane-half selects K pair {0,1} vs {2,3}
    const bool nvalid = (n < 9);

    // B matrix (4x16): VGPR0 = K(4c+2kh), VGPR1 = K(4c+2kh+1), col = n
    float bw[3][2];
#pragma unroll
    for (int c = 0; c < 3; ++c) {
        bw[c][0] = nvalid ? act_W[n * 12 + 4 * c + 2 * kh + 0] : 0.f;
        bw[c][1] = nvalid ? act_W[n * 12 + 4 * c + 2 * kh + 1] : 0.f;
    }
    const float bias = nvalid ? act_b[n] : 0.f;

#pragma unroll
    for (int tile = 0; tile < 2; ++tile) {
        const int rb = wv * 32 + tile * 16;  // 16 batch rows of this tile
        v8f acc = {};
#pragma unroll
        for (int c = 0; c < 3; ++c) {
            v2f a, b;
            // A (16x4): row = rb + n (both lane halves cover M=0..15)
            a.x = s_flat[(rb + n) * FSTRIDE + 4 * c + 2 * kh + 0];
            a.y = s_flat[(rb + n) * FSTRIDE + 4 * c + 2 * kh + 1];
            b.x = bw[c][0];
            b.y = bw[c][1];
            acc = __builtin_amdgcn_wmma_f32_16x16x4_f32(
                /*neg_a=*/false, a, /*neg_b=*/false, b,
                /*c_mod=*/(short)0, acc, /*reuse_a=*/false, /*reuse_b=*/false);
        }
        // D (16x16 f32): VGPR j -> row (j + 8*kh), col n. Park in LDS; no
        // divergent global stores here (EXEC restored before next WMMA).
        if (nvalid) {
#pragma unroll
            for (int j = 0; j < 8; ++j)
                s_q[(rb + j + 8 * kh) * 9 + n] = acc[j] + bias;
        }
    }
    __syncthreads();

    // ---- Coalesced output: 256 rows x 9 floats, stride-BLK streaming ----
    float* gout = out + blockBase * 9;
    if (fullBlock) {
#pragma unroll
        for (int k = 0; k < 9; ++k)
            gout[(size_t)k * BLK + t] = s_q[k * BLK + t];
    } else {
        const size_t remain = (size_t)Bn * 9 - blockBase * 9;
#pragma unroll
        for (int k = 0; k < 9; ++k) {
            size_t idx = (size_t)k * BLK + t;
            if (idx < remain) gout[idx] = s_q[k * BLK + t];
        }
    }
}

extern "C" void kernel_launch(void* const* d_in, const int* in_sizes, int n_in,
                              void* d_out, int out_size, void* d_ws, size_t ws_size,
                              hipStream_t stream) {
    const float* state       = (const float*)d_in[0];
    const float* c_templates = (const float*)d_in[1];
    const float* c_gammas    = (const float*)d_in[2];
    const float* a_templates = (const float*)d_in[3];
    const float* a_gammas    = (const float*)d_in[4];
    const float* a_body_W    = (const float*)d_in[5];
    const float* a_body_b    = (const float*)d_in[6];
    const float* a_head_W    = (const float*)d_in[7];
    const float* a_head_b    = (const float*)d_in[8];
    const float* act_W       = (const float*)d_in[9];
    const float* act_b       = (const float*)d_in[10];

    const int Bn = in_sizes[0] / 18;
    const int blocks = (Bn + BLK - 1) / BLK;

    logic_net_wmma_kernel<<<blocks, BLK, 0, stream>>>(
        state, c_templates, c_gammas, a_templates, a_gammas,
        a_body_W, a_body_b, a_head_W, a_head_b, act_W, act_b,
        (float*)d_out, Bn);
}